// GNNNodeEmbedding_79388175499492
// MI455X (gfx1250) — compile-verified
//
#include <hip/hip_runtime.h>
#include <cstdint>
#include <cstddef>

// CDNA5 / gfx1250. wave32.
typedef __attribute__((ext_vector_type(2))) float v2f;
typedef __attribute__((ext_vector_type(8))) float v8f;

#define F_IN  100
#define F_MID 64
#define F_OUT 32

__device__ __forceinline__ void atomic_add_f32(float* p, float v) {
  // HW f32 atomic (global_atomic_add_f32); agg buffers are device-coarse memory.
  unsafeAtomicAdd(p, v);
}

// ---------------------------------------------------------------------------
// init: deg=1.0 (self loop), agg1 = 0, out = 0
// ---------------------------------------------------------------------------
__global__ void init_kernel(float* __restrict__ deg, float* __restrict__ agg1,
                            float* __restrict__ out, int nNodes) {
  int i = blockIdx.x * blockDim.x + threadIdx.x;
  if (i < nNodes) deg[i] = 1.0f;
  if (i < nNodes * F_MID) agg1[i] = 0.0f;
  if (i < nNodes * F_OUT) out[i] = 0.0f;
}

// in-degree over real edges (dst half of edge_index, int64)
__global__ void deg_kernel(const long long* __restrict__ ei, float* __restrict__ deg, int E) {
  int e = blockIdx.x * blockDim.x + threadIdx.x;
  if (e < E) {
    int d = (int)ei[(size_t)E + e];
    atomic_add_f32(&deg[d], 1.0f);
  }
}

__global__ void dinv_kernel(float* __restrict__ deg, int n) {
  int i = blockIdx.x * blockDim.x + threadIdx.x;
  if (i < n) {
    float d = deg[i];
    deg[i] = (d > 0.0f) ? rsqrtf(d) : 0.0f;
  }
}

// ---------------------------------------------------------------------------
// GEMM via V_WMMA_F32_16X16X4_F32.  H[nRows,N] = A[nRows,K] @ W[K,N]
// 256 threads = 8 waves; each wave computes a 16 x (NT*16) tile.
//
// W is staged in LDS *swizzled by k-pairs*: lWs[((k/2)*N + n)*2 + {0,1}] =
// W[k][n], W[k+1][n].  The B fragment for a 16x16x4 step needs VGPR pair
// (K=kb, K=kb+1) per lane -> with this layout it is ONE aligned ds_load_b64
// into an even VGPR pair (no repacking moves).
//
// A-frag  (16x4 f32): lane<16 -> M=lane, VGPR{0,1}=K{0,1}; lane>=16 -> K{2,3}.
// B-frag  (4x16 f32): VGPR v, lanes 0-15 -> K=v, lanes 16-31 -> K=v+2, N=lane&15.
// C/D     (16x16 f32): VGPR v, lanes 0-15 -> M=v, lanes 16-31 -> M=v+8.
// ---------------------------------------------------------------------------
template <int NT>
__global__ void gemm_wmma_kernel(const float* __restrict__ A, const float* __restrict__ W,
                                 float* __restrict__ H, int nRows, int K, int N) {
  __shared__ float lWs[F_IN * F_MID];  // 6400 floats (25.6 KB), covers both layers
  const int tid = threadIdx.x;
  const int nPairs = (K >> 1) * N;
  for (int p = tid; p < nPairs; p += blockDim.x) {
    const int k = (p / N) << 1;
    const int n = p - (p / N) * N;
    lWs[p * 2 + 0] = W[k * N + n];
    lWs[p * 2 + 1] = W[(k + 1) * N + n];
  }
  __syncthreads();

  const int wave = tid >> 5;
  const int lane = tid & 31;
  const int half = lane >> 4;  // 0: lanes 0-15, 1: lanes 16-31
  const int m    = lane & 15;

  const int rowBase = blockIdx.x * ((int)blockDim.x >> 5) * 16 + wave * 16;
  if (rowBase >= nRows) return;  // wave-uniform; EXEC stays all-ones for WMMA

  v8f c[NT];
#pragma unroll
  for (int t = 0; t < NT; ++t) c[t] = {};

  const float* arow = A + (size_t)(rowBase + m) * K + half * 2;
  // B pair base for this lane: pair index = ks*2 + half, column = t*16 + m
  const float* brow = lWs + ((half * N + m) << 1);
  const int kSteps = K >> 2;
  for (int ks = 0; ks < kSteps; ++ks) {
    const v2f a = *(const v2f*)(arow + ks * 4);  // K = ks*4 + half*2 + {0,1}
#pragma unroll
    for (int t = 0; t < NT; ++t) {
      const v2f b = *(const v2f*)(brow + ((ks * 2 * N + t * 16) << 1));
      c[t] = __builtin_amdgcn_wmma_f32_16x16x4_f32(
          /*neg_a=*/false, a, /*neg_b=*/false, b,
          /*c_mod=*/(short)0, c[t], /*reuse_a=*/false, /*reuse_b=*/false);
    }
  }

#pragma unroll
  for (int t = 0; t < NT; ++t) {
    const int col = t * 16 + m;
#pragma unroll
    for (int v = 0; v < 8; ++v) {
      const int row = rowBase + half * 8 + v;
      H[(size_t)row * N + col] = c[t][v];
    }
  }
}

// ---------------------------------------------------------------------------
// Edge scatter, layer 1 (F=64): one wave per edge, 2 features per lane.
// h1/agg1 are ~25.6 MB each -> resident in the 192 MB L2; atomics resolve in L2.
// ---------------------------------------------------------------------------
__global__ void scatter64_kernel(const long long* __restrict__ ei,
                                 const float* __restrict__ dinv,
                                 const float* __restrict__ h,
                                 float* __restrict__ agg, int E) {
  const int t = blockIdx.x * blockDim.x + threadIdx.x;
  const int e = t >> 5;
  const int lane = t & 31;
  if (e >= E) return;
  const int s = (int)ei[e];
  const int d = (int)ei[(size_t)E + e];
  const float nrm = dinv[s] * dinv[d];
  const v2f hv = *(const v2f*)(h + (size_t)s * F_MID + lane * 2);
  float* o = agg + (size_t)d * F_MID + lane * 2;
  atomic_add_f32(o, hv.x * nrm);
  atomic_add_f32(o + 1, hv.y * nrm);
}

// Edge scatter, layer 2 (F=32): one wave per edge, 1 feature per lane.
__global__ void scatter32_kernel(const long long* __restrict__ ei,
                                 const float* __restrict__ dinv,
                                 const float* __restrict__ h,
                                 float* __restrict__ agg, int E) {
  const int t = blockIdx.x * blockDim.x + threadIdx.x;
  const int e = t >> 5;
  const int lane = t & 31;
  if (e >= E) return;
  const int s = (int)ei[e];
  const int d = (int)ei[(size_t)E + e];
  const float nrm = dinv[s] * dinv[d];
  atomic_add_f32(agg + (size_t)d * F_OUT + lane,
                 h[(size_t)s * F_OUT + lane] * nrm);
}

// z1 = relu(agg1 + h1*dinv^2 (self loop) + b1)   — in place on agg1 buffer
__global__ void post1_kernel(const float* __restrict__ h1, const float* __restrict__ dinv,
                             const float* __restrict__ b1, float* __restrict__ z,
                             int nNodes) {
  const int i = blockIdx.x * blockDim.x + threadIdx.x;
  if (i >= nNodes * F_MID) return;
  const int node = i >> 6;
  const int f = i & (F_MID - 1);
  const float di = dinv[node];
  const float v = z[i] + h1[i] * di * di + b1[f];
  z[i] = fmaxf(v, 0.0f);
}

// out = agg2 + h2*dinv^2 (self loop) + b2
__global__ void post2_kernel(const float* __restrict__ h2, const float* __restrict__ dinv,
                             const float* __restrict__ b2, float* __restrict__ out,
                             int nNodes) {
  const int i = blockIdx.x * blockDim.x + threadIdx.x;
  if (i >= nNodes * F_OUT) return;
  const int node = i >> 5;
  const int f = i & (F_OUT - 1);
  const float di = dinv[node];
  out[i] = out[i] + h2[i] * di * di + b2[f];
}

// ---------------------------------------------------------------------------
extern "C" void kernel_launch(void* const* d_in, const int* in_sizes, int n_in,
                              void* d_out, int out_size, void* d_ws, size_t ws_size,
                              hipStream_t stream) {
  const float*     x  = (const float*)d_in[0];
  const long long* ei = (const long long*)d_in[1];  // [2, E] int64
  const float*     W1 = (const float*)d_in[2];
  const float*     b1 = (const float*)d_in[3];
  const float*     W2 = (const float*)d_in[4];
  const float*     b2 = (const float*)d_in[5];
  float* out = (float*)d_out;

  const int nNodes = in_sizes[0] / F_IN;  // 100000
  const int E      = in_sizes[1] / 2;     // 1600000

  // Workspace layout (floats): [dinv | h1(+h2 reuse) | agg1/z1]
  float* ws   = (float*)d_ws;
  float* dinv = ws;                                // nNodes (deg then dinv, in place)
  float* h1   = ws + 131072;                       // nNodes*64
  float* z1   = h1 + (size_t)nNodes * F_MID;       // nNodes*64 (agg1 -> z1 in place)
  float* h2   = h1;                                // h1 dead after post1; reuse buffer

  const int TB = 256;

  // 1) init deg/agg1/out
  {
    int tot = nNodes * F_MID;  // largest of the three extents
    init_kernel<<<(tot + TB - 1) / TB, TB, 0, stream>>>(dinv, z1, out, nNodes);
  }
  // 2) degree + rsqrt
  deg_kernel<<<(E + TB - 1) / TB, TB, 0, stream>>>(ei, dinv, E);
  dinv_kernel<<<(nNodes + TB - 1) / TB, TB, 0, stream>>>(dinv, nNodes);

  // 3) h1 = x @ W1   (WMMA fp32; 8 waves/block -> 128 rows/block)
  const int gemmBlocks = (nNodes + 127) / 128;
  gemm_wmma_kernel<4><<<gemmBlocks, TB, 0, stream>>>(x, W1, h1, nNodes, F_IN, F_MID);

  // 4) layer-1 edge aggregation (wave per edge)
  {
    size_t th = (size_t)E * 32;
    scatter64_kernel<<<(unsigned)((th + TB - 1) / TB), TB, 0, stream>>>(ei, dinv, h1, z1, E);
  }
  // 5) self-loop + bias + relu
  post1_kernel<<<(nNodes * F_MID + TB - 1) / TB, TB, 0, stream>>>(h1, dinv, b1, z1, nNodes);

  // 6) h2 = z1 @ W2
  gemm_wmma_kernel<2><<<gemmBlocks, TB, 0, stream>>>(z1, W2, h2, nNodes, F_MID, F_OUT);

  // 7) layer-2 edge aggregation into out
  {
    size_t th = (size_t)E * 32;
    scatter32_kernel<<<(unsigned)((th + TB - 1) / TB), TB, 0, stream>>>(ei, dinv, h2, out, E);
  }
  // 8) self-loop + bias
  post2_kernel<<<(nNodes * F_OUT + TB - 1) / TB, TB, 0, stream>>>(h2, dinv, b2, out, nNodes);
}